// HOIST_58136677318718
// MI455X (gfx1250) — compile-verified
//
#include <hip/hip_runtime.h>
#include <math.h>

#define N_    2000
#define T_    64
#define DDYN_ 64
#define RNN_  128
#define HID_  16
#define DIS_  4

typedef __attribute__((ext_vector_type(2))) float v2f;
typedef __attribute__((ext_vector_type(8))) float v8f;

__device__ __forceinline__ float sigmoidf_(float x){ return 1.0f/(1.0f+expf(-x)); }
__device__ __forceinline__ float leakyf_(float x){ return x>0.0f ? x : 0.01f*x; }
__device__ __forceinline__ float geluf_(float x){ return 0.5f*x*(1.0f+erff(x*0.70710678118654752f)); }

// D = A(16x4) * B(4x16) + C, fp32 WMMA (codegen-verified this target):
// (neg_a, A, neg_b, B, c_mod, C, reuse_a, reuse_b)
__device__ __forceinline__ v8f wmma4(v2f a, v2f b, v8f c){
  return __builtin_amdgcn_wmma_f32_16x16x4_f32(false, a, false, b, (short)0, c, false, false);
}

// CDNA5 async global->LDS copy (ASYNCcnt-tracked). 16B per lane per issue.
__device__ __forceinline__ void async_b128_to_lds(const float* gptr, float* lptr){
  unsigned lds = (unsigned)(size_t)lptr;               // flat low 32 bits == LDS offset
  unsigned long long ga = (unsigned long long)gptr;
  asm volatile("global_load_async_to_lds_b128 %0, %1, off"
               :: "v"(lds), "v"(ga) : "memory");
}
__device__ __forceinline__ void wait_async_le(int n){
  if (n == 0) asm volatile("s_wait_asynccnt 0x0" ::: "memory");
  else        asm volatile("s_wait_asynccnt 0x1" ::: "memory");
}

// ---------------------------------------------------------------------------
// prep: wa1[k][d] = sum_e w_list[k,d,e]*a_list[k,e,0]
//       wa2[k][d] = sum_e w_list[k,d,e]*a_list[k,16+e,0]
//       wadis[d]  = sum_e W_dis[d,e]*a_dis[e,0]
// ---------------------------------------------------------------------------
__global__ void k_prep(const float* __restrict__ w_list, const float* __restrict__ a_list,
                       const float* __restrict__ W_dis, const float* __restrict__ a_dis,
                       float* __restrict__ wa1, float* __restrict__ wa2,
                       float* __restrict__ wadis){
  int tid = threadIdx.x;
  if (tid < 112){
    int k = tid >> 4, d = tid & 15;
    float s1 = 0.f, s2 = 0.f;
    for (int e = 0; e < 16; e++){
      float w = w_list[(k*16 + d)*16 + e];
      s1 += w * a_list[k*32 + e];
      s2 += w * a_list[k*32 + 16 + e];
    }
    wa1[tid] = s1; wa2[tid] = s2;
  }
  if (tid < 4){
    float s = 0.f;
    for (int e = 0; e < 4; e++) s += W_dis[tid*4 + e] * a_dis[e];
    wadis[tid] = s;
  }
}

// ---------------------------------------------------------------------------
// per-(n,t91) type reductions: vac(34), hos(4), claim(20) -> scalar series
// ---------------------------------------------------------------------------
__global__ void k_typered(const float* __restrict__ vac, const float* __restrict__ hos,
                          const float* __restrict__ claim,
                          const float* __restrict__ vtW, const float* __restrict__ vtB,
                          const float* __restrict__ htW, const float* __restrict__ htB,
                          const float* __restrict__ ctW, const float* __restrict__ ctB,
                          float* __restrict__ vac_s, float* __restrict__ hos_s,
                          float* __restrict__ claim_s){
  int id = blockIdx.x*blockDim.x + threadIdx.x;
  if (id >= N_*91) return;
  float s = vtB[0];
  const float* vp = vac + (size_t)id*34;
  for (int j = 0; j < 34; j++) s += vp[j]*vtW[j];
  vac_s[id] = s;
  s = htB[0];
  const float* hp = hos + (size_t)id*4;
  for (int j = 0; j < 4; j++) s += hp[j]*htW[j];
  hos_s[id] = s;
  s = ctB[0];
  const float* cp = claim + (size_t)id*20;
  for (int j = 0; j < 20; j++) s += cp[j]*ctW[j];
  claim_s[id] = s;
}

// ---------------------------------------------------------------------------
// feats[(k*N+n)*16+o] for k = 0..6
// ---------------------------------------------------------------------------
__global__ void k_feats(const float* __restrict__ county, const float* __restrict__ covid,
                        const float* __restrict__ vac_s, const float* __restrict__ hos_s,
                        const float* __restrict__ claim_s, const float* __restrict__ cte,
                        const int* __restrict__ month,
                        const float* __restrict__ countyW, const float* __restrict__ countyB,
                        const float* __restrict__ covidW,  const float* __restrict__ covidB,
                        const float* __restrict__ vacW,    const float* __restrict__ vacB,
                        const float* __restrict__ hosW,    const float* __restrict__ hosB,
                        const float* __restrict__ claimW,  const float* __restrict__ claimB,
                        const float* __restrict__ cteW,    const float* __restrict__ cteB,
                        const float* __restrict__ month_table,
                        float* __restrict__ feats){
  int id = blockIdx.x*blockDim.x + threadIdx.x;
  if (id >= 7*N_*HID_) return;
  int k = id / (N_*HID_);
  int rem = id % (N_*HID_);
  int n = rem / HID_, o = rem % HID_;
  float s = 0.f;
  switch (k){
    case 0: s = countyB[o]; for (int i=0;i<14;i++) s += county[n*14+i]*countyW[o*14+i]; break;
    case 1: s = covidB[o];  for (int i=0;i<91;i++) s += covid[n*91+i]*covidW[o*91+i];   break;
    case 2: s = vacB[o];    for (int i=0;i<91;i++) s += vac_s[n*91+i]*vacW[o*91+i];     break;
    case 3: s = hosB[o];    for (int i=0;i<91;i++) s += hos_s[n*91+i]*hosW[o*91+i];     break;
    case 4: s = claimB[o];  for (int i=0;i<91;i++) s += claim_s[n*91+i]*claimW[o*91+i]; break;
    case 5: s = cteB[o];    for (int i=0;i<3;i++)  s += cte[n*3+i]*cteW[o*3+i];         break;
    case 6: s = month_table[month[n]*HID_ + o]; break;
  }
  feats[id] = s;
}

// ---------------------------------------------------------------------------
// static_emb = gelu(concat feats) @ static_W^T + b   (112 -> 16)
// ---------------------------------------------------------------------------
__global__ void k_semb(const float* __restrict__ feats, const float* __restrict__ sW,
                       const float* __restrict__ sB, float* __restrict__ semb){
  int id = blockIdx.x*blockDim.x + threadIdx.x;
  if (id >= N_*HID_) return;
  int n = id / HID_, o = id % HID_;
  float s = sB[o];
  for (int k = 0; k < 7; k++)
    for (int d = 0; d < 16; d++)
      s += geluf_(feats[(k*N_ + n)*16 + d]) * sW[o*112 + k*16 + d];
  semb[id] = s;
}

// ---------------------------------------------------------------------------
// l[k,n] = feats[k,n,:] . wa1[k], r[k,n] = feats[k,n,:] . wa2[k]
// ---------------------------------------------------------------------------
__global__ void k_lr(const float* __restrict__ feats, const float* __restrict__ wa1,
                     const float* __restrict__ wa2, float* __restrict__ l,
                     float* __restrict__ r){
  int id = blockIdx.x*blockDim.x + threadIdx.x;
  if (id >= 7*N_) return;
  int k = id / N_, n = id % N_;
  float s1 = 0.f, s2 = 0.f;
  for (int d = 0; d < 16; d++){
    float f = feats[(k*N_ + n)*16 + d];
    s1 += f*wa1[k*16 + d];
    s2 += f*wa2[k*16 + d];
  }
  l[id] = s1; r[id] = s2;
}

// ---------------------------------------------------------------------------
// static_dis row: softmax_m( sum_k sig(l[k,n]+r[k,m]) + sig(dist[n,m,:].wadis) )
// ---------------------------------------------------------------------------
__global__ void __launch_bounds__(256) k_sdrow(const float* __restrict__ l,
                                               const float* __restrict__ r,
                                               const float* __restrict__ dist,
                                               const float* __restrict__ wadis,
                                               float* __restrict__ sd){
  __shared__ float red[256];
  int n = blockIdx.x, tid = threadIdx.x;
  float ln[7];
  for (int k = 0; k < 7; k++) ln[k] = l[k*N_ + n];
  float w0 = wadis[0], w1 = wadis[1], w2 = wadis[2], w3 = wadis[3];
  float sreg[8];
  for (int i = 0; i < 8; i++){
    int m = tid + i*256;
    float s = -1e30f;
    if (m < N_){
      s = 0.f;
      for (int k = 0; k < 7; k++) s += sigmoidf_(ln[k] + r[k*N_ + m]);
      const float* dp = dist + ((size_t)n*N_ + m)*DIS_;
      s += sigmoidf_(dp[0]*w0 + dp[1]*w1 + dp[2]*w2 + dp[3]*w3);
    }
    sreg[i] = s;
  }
  float mx = sreg[0];
  for (int i = 1; i < 8; i++) mx = fmaxf(mx, sreg[i]);
  red[tid] = mx; __syncthreads();
  for (int off = 128; off > 0; off >>= 1){
    if (tid < off) red[tid] = fmaxf(red[tid], red[tid+off]);
    __syncthreads();
  }
  mx = red[0]; __syncthreads();
  float ssum = 0.f;
  for (int i = 0; i < 8; i++){
    int m = tid + i*256;
    if (m < N_){ sreg[i] = expf(sreg[i]-mx); ssum += sreg[i]; }
  }
  red[tid] = ssum; __syncthreads();
  for (int off = 128; off > 0; off >>= 1){
    if (tid < off) red[tid] += red[tid+off];
    __syncthreads();
  }
  float inv = 1.0f/red[0];
  for (int i = 0; i < 8; i++){
    int m = tid + i*256;
    if (m < N_) sd[(size_t)n*N_ + m] = sreg[i]*inv;
  }
}

// ---------------------------------------------------------------------------
// cur_weight MLP: 64 -> leaky -> 128 -> sigmoid -> 64, fused reweight.
// One wave per block, 16 flattened (n,t) rows, WMMA f32 16x16x4.
// Row tile staged via async global->LDS (B128 per lane).
// ---------------------------------------------------------------------------
__global__ void __launch_bounds__(32) k_mlp(const float* __restrict__ dyn,
                                            const float* __restrict__ dw1W,
                                            const float* __restrict__ dw1B,
                                            const float* __restrict__ dw2W,
                                            const float* __restrict__ dw2B,
                                            float* __restrict__ cw_out,
                                            float* __restrict__ x_out){
  __shared__ float vrow[16][64];
  __shared__ float arow[16][128];
  int lane = threadIdx.x;
  int r0 = blockIdx.x * 16;
  int m = lane & 15, hi = lane >> 4;
  // 16x64 floats = 256 lanes-worth of B128 -> 8 issues per lane
  #pragma unroll
  for (int j = 0; j < 8; j++){
    int f = (j*32 + lane) * 4;            // 16B-aligned flat offset in tile
    async_b128_to_lds(dyn + (size_t)r0*64 + f, &vrow[f>>6][f&63]);
  }
  wait_async_le(0);
  __syncthreads();
  // layer 1: z[16x128] tiles
  #pragma unroll
  for (int tt = 0; tt < 8; tt++){
    int col = tt*16 + m;
    v8f acc; float bb = dw1B[col];
    #pragma unroll
    for (int j = 0; j < 8; j++) acc[j] = bb;
    const float* wr = dw1W + (size_t)col*64;
    #pragma unroll
    for (int k0 = 0; k0 < 64; k0 += 4){
      int kk = k0 + 2*hi;
      v2f a, b;
      a.x = vrow[m][kk]; a.y = vrow[m][kk+1];
      b.x = wr[kk];      b.y = wr[kk+1];
      acc = wmma4(a, b, acc);
    }
    #pragma unroll
    for (int j = 0; j < 8; j++) arow[j + 8*hi][col] = leakyf_(acc[j]);
  }
  __syncthreads();
  // layer 2: w[16x64] tiles, sigmoid, reweight
  #pragma unroll
  for (int tt = 0; tt < 4; tt++){
    int col = tt*16 + m;
    v8f acc; float bb = dw2B[col];
    #pragma unroll
    for (int j = 0; j < 8; j++) acc[j] = bb;
    const float* wr = dw2W + (size_t)col*128;
    #pragma unroll
    for (int k0 = 0; k0 < 128; k0 += 4){
      int kk = k0 + 2*hi;
      v2f a, b;
      a.x = arow[m][kk]; a.y = arow[m][kk+1];
      b.x = wr[kk];      b.y = wr[kk+1];
      acc = wmma4(a, b, acc);
    }
    #pragma unroll
    for (int j = 0; j < 8; j++){
      int row = j + 8*hi;
      float w = sigmoidf_(acc[j]);
      cw_out[(size_t)(r0+row)*64 + col] = w;
      x_out[(size_t)(r0+row)*64 + col]  = w * vrow[row][col];
    }
  }
}

// ---------------------------------------------------------------------------
// LSTM: one block per 16 sequences, 8 waves; WMMA f32 16x16x4 gate GEMM.
// Wih B-fragments hoisted to registers (t-invariant, 128 VGPRs/lane);
// x tiles double-buffered via async global->LDS; Whh streamed through cache.
// ---------------------------------------------------------------------------
__global__ void __launch_bounds__(256) k_lstm(const float* __restrict__ x,
                                              const float* __restrict__ Wih,
                                              const float* __restrict__ Whh,
                                              const float* __restrict__ bih,
                                              const float* __restrict__ bhh,
                                              float* __restrict__ h_all,
                                              float* __restrict__ hT,
                                              float* __restrict__ cT){
  __shared__ float xs[2][16][64]; //  8 KB (double-buffered async tile)
  __shared__ float hs[16][128];   //  8 KB
  __shared__ float cs[16][128];   //  8 KB
  __shared__ float zs[16][512];   // 32 KB
  int tid = threadIdx.x;
  int lane = tid & 31, wv = tid >> 5;
  int m = lane & 15, hi = lane >> 4;
  int n0 = blockIdx.x * 16;
  // async x-tile stage: one B128 per thread covers the 16x64 tile
  int xrow = tid >> 4;
  int xcol = (tid & 15) * 4;
  const float* xbase = x + (size_t)(n0 + xrow)*T_*64 + xcol;

  // hoist t-invariant Wih fragments: 4 tiles x 16 K-chunks x v2f = 128 VGPRs
  v2f bx[4][16];
  #pragma unroll
  for (int tt = 0; tt < 4; tt++){
    int col = wv*64 + tt*16 + m;
    const float* wr = Wih + (size_t)col*64;
    #pragma unroll
    for (int c = 0; c < 16; c++){
      int kk = c*4 + 2*hi;
      bx[tt][c].x = wr[kk]; bx[tt][c].y = wr[kk+1];
    }
  }
  for (int i = tid; i < 16*128; i += 256){ hs[i>>7][i&127] = 0.f; cs[i>>7][i&127] = 0.f; }
  async_b128_to_lds(xbase, &xs[0][xrow][xcol]);   // prime t=0
  __syncthreads();

  for (int t = 0; t < T_; t++){
    if (t + 1 < T_){
      async_b128_to_lds(xbase + (size_t)(t+1)*64, &xs[(t+1)&1][xrow][xcol]);
      wait_async_le(1);   // in-order: tile t has landed
    } else {
      wait_async_le(0);
    }
    __syncthreads();
    const float (*xt)[64] = xs[t & 1];
    // each wave: 4 gate tiles (64 gate columns), K = 64 (x, reg B) + 128 (h)
    #pragma unroll
    for (int tt = 0; tt < 4; tt++){
      int col = wv*64 + tt*16 + m;
      v8f acc; float bb = bih[col] + bhh[col];
      #pragma unroll
      for (int j = 0; j < 8; j++) acc[j] = bb;
      #pragma unroll
      for (int c = 0; c < 16; c++){
        int kk = c*4 + 2*hi;
        v2f a;
        a.x = xt[m][kk]; a.y = xt[m][kk+1];
        acc = wmma4(a, bx[tt][c], acc);
      }
      const float* wh = Whh + (size_t)col*128;
      #pragma unroll
      for (int k0 = 0; k0 < 128; k0 += 4){
        int kk = k0 + 2*hi;
        v2f a, b;
        a.x = hs[m][kk]; a.y = hs[m][kk+1];
        b.x = wh[kk];    b.y = wh[kk+1];
        acc = wmma4(a, b, acc);
      }
      #pragma unroll
      for (int j = 0; j < 8; j++) zs[j + 8*hi][col] = acc[j];
    }
    __syncthreads();
    for (int e = tid; e < 16*128; e += 256){
      int row = e>>7, rr = e&127;
      float zi = zs[row][rr],       zf = zs[row][128+rr];
      float zg = zs[row][256+rr],   zo = zs[row][384+rr];
      float c = sigmoidf_(zf)*cs[row][rr] + sigmoidf_(zi)*tanhf(zg);
      float h = sigmoidf_(zo)*tanhf(c);
      cs[row][rr] = c; hs[row][rr] = h;
      h_all[((size_t)(n0+row)*T_ + t)*128 + rr] = h;
    }
    __syncthreads();
  }
  for (int e = tid; e < 16*128; e += 256){
    int row = e>>7, rr = e&127;
    hT[(size_t)(n0+row)*128 + rr] = hs[row][rr];
    cT[(size_t)(n0+row)*128 + rr] = cs[row][rr];
  }
}

// ---------------------------------------------------------------------------
// last_h[n] = h[n,idx] + sum_m sd[n,m]*h[m,idx]; then 144->128 leaky ->4 head
// ---------------------------------------------------------------------------
__global__ void __launch_bounds__(128) k_head(const int* __restrict__ mask,
                                              const float* __restrict__ sd,
                                              const float* __restrict__ h_all,
                                              const float* __restrict__ semb,
                                              const float* __restrict__ linW,
                                              const float* __restrict__ linB,
                                              const float* __restrict__ lin2W,
                                              const float* __restrict__ lin2B,
                                              float* __restrict__ y){
  __shared__ float sdrow[128];
  __shared__ float lasth[128];
  __shared__ float se[16];
  __shared__ float u[128];
  __shared__ int tshare;
  int n = blockIdx.x, tid = threadIdx.x;
  if (tid == 0){
    int s = 0;
    for (int t = 0; t < T_; t++) s += mask[n*T_ + t];
    tshare = s - 1;
  }
  if (tid < 16) se[tid] = semb[n*16 + tid];
  __syncthreads();
  int tn = tshare;
  float acc = h_all[((size_t)n*T_ + tn)*128 + tid];
  for (int base = 0; base < N_; base += 128){
    int mm = base + tid;
    sdrow[tid] = (mm < N_) ? sd[(size_t)n*N_ + mm] : 0.f;
    __syncthreads();
    int lim = (N_ - base) < 128 ? (N_ - base) : 128;
    for (int j = 0; j < lim; j++)
      acc += sdrow[j] * h_all[((size_t)(base+j)*T_ + tn)*128 + tid];
    __syncthreads();
  }
  lasth[tid] = acc;
  __syncthreads();
  float s = linB[tid];
  for (int j = 0; j < 128; j++) s += lasth[j]*linW[tid*144 + j];
  for (int j = 0; j < 16;  j++) s += se[j]*linW[tid*144 + 128 + j];
  u[tid] = leakyf_(s);
  __syncthreads();
  if (tid < 4){
    float yy = lin2B[tid];
    for (int j = 0; j < 128; j++) yy += u[j]*lin2W[tid*128 + j];
    y[(size_t)n*4 + tid] = yy;
  }
}

// ---------------------------------------------------------------------------
extern "C" void kernel_launch(void* const* d_in, const int* in_sizes, int n_in,
                              void* d_out, int out_size, void* d_ws, size_t ws_size,
                              hipStream_t stream){
  (void)in_sizes; (void)n_in; (void)out_size; (void)ws_size;
  const float* dynamic0 = (const float*)d_in[0];
  const int*   mask     = (const int*)  d_in[1];
  const float* county   = (const float*)d_in[2];
  const float* covid    = (const float*)d_in[3];
  const float* vac      = (const float*)d_in[4];
  const float* hos      = (const float*)d_in[5];
  const float* claim    = (const float*)d_in[6];
  const float* cte      = (const float*)d_in[7];
  const int*   month    = (const int*)  d_in[8];
  const float* distance = (const float*)d_in[9];
  const float* county_W = (const float*)d_in[10];
  const float* county_b = (const float*)d_in[11];
  const float* covid_W  = (const float*)d_in[12];
  const float* covid_b  = (const float*)d_in[13];
  const float* vact_W   = (const float*)d_in[14];
  const float* vact_b   = (const float*)d_in[15];
  const float* vactm_W  = (const float*)d_in[16];
  const float* vactm_b  = (const float*)d_in[17];
  const float* host_W   = (const float*)d_in[18];
  const float* host_b   = (const float*)d_in[19];
  const float* hostm_W  = (const float*)d_in[20];
  const float* hostm_b  = (const float*)d_in[21];
  const float* clt_W    = (const float*)d_in[22];
  const float* clt_b    = (const float*)d_in[23];
  const float* cltm_W   = (const float*)d_in[24];
  const float* cltm_b   = (const float*)d_in[25];
  const float* cte_W    = (const float*)d_in[26];
  const float* cte_b    = (const float*)d_in[27];
  const float* month_tb = (const float*)d_in[28];
  const float* static_W = (const float*)d_in[29];
  const float* static_b = (const float*)d_in[30];
  const float* w_list   = (const float*)d_in[31];
  const float* a_list   = (const float*)d_in[32];
  const float* W_dis    = (const float*)d_in[33];
  const float* a_dis    = (const float*)d_in[34];
  const float* dw1_W    = (const float*)d_in[35];
  const float* dw1_b    = (const float*)d_in[36];
  const float* dw2_W    = (const float*)d_in[37];
  const float* dw2_b    = (const float*)d_in[38];
  const float* Wih      = (const float*)d_in[39];
  const float* Whh      = (const float*)d_in[40];
  const float* bih      = (const float*)d_in[41];
  const float* bhh      = (const float*)d_in[42];
  const float* lin_W    = (const float*)d_in[43];
  const float* lin_b    = (const float*)d_in[44];
  const float* lin2_W   = (const float*)d_in[45];
  const float* lin2_b   = (const float*)d_in[46];

  // output layout: y | static_dis | cur_weight | hT | cT
  float* out    = (float*)d_out;
  float* out_y  = out;
  float* out_sd = out + 8000;
  float* out_cw = out + 4008000;
  float* out_hT = out + 12200000;
  float* out_cT = out + 12456000;

  // workspace layout (floats)
  float* ws      = (float*)d_ws;
  float* vac_s   = ws;                    // 182000
  float* hos_s   = vac_s   + 182000;      // 182000
  float* claim_s = hos_s   + 182000;      // 182000
  float* feats   = claim_s + 182000;      // 224000 (7,N,16)
  float* semb    = feats   + 224000;      // 32000
  float* lbuf    = semb    + 32000;       // 14000  (7,N)
  float* rbuf    = lbuf    + 14000;       // 14000
  float* wa1     = rbuf    + 14000;       // 112
  float* wa2     = wa1     + 112;         // 112
  float* wadis   = wa2     + 112;         // 16 (padded, keeps xw 16B-aligned)
  float* xw      = wadis   + 16;          // 8192000 (N,T,64)
  float* h_all   = xw      + 8192000;     // 16384000 (N,T,128)

  k_prep<<<1, 128, 0, stream>>>(w_list, a_list, W_dis, a_dis, wa1, wa2, wadis);
  k_typered<<<(N_*91 + 255)/256, 256, 0, stream>>>(vac, hos, claim,
      vact_W, vact_b, host_W, host_b, clt_W, clt_b, vac_s, hos_s, claim_s);
  k_feats<<<(7*N_*HID_ + 255)/256, 256, 0, stream>>>(county, covid, vac_s, hos_s,
      claim_s, cte, month, county_W, county_b, covid_W, covid_b,
      vactm_W, vactm_b, hostm_W, hostm_b, cltm_W, cltm_b, cte_W, cte_b,
      month_tb, feats);
  k_semb<<<(N_*HID_ + 255)/256, 256, 0, stream>>>(feats, static_W, static_b, semb);
  k_lr<<<(7*N_ + 255)/256, 256, 0, stream>>>(feats, wa1, wa2, lbuf, rbuf);
  k_sdrow<<<N_, 256, 0, stream>>>(lbuf, rbuf, distance, wadis, out_sd);
  k_mlp<<<(N_*T_)/16, 32, 0, stream>>>(dynamic0, dw1_W, dw1_b, dw2_W, dw2_b,
      out_cw, xw);
  k_lstm<<<N_/16, 256, 0, stream>>>(xw, Wih, Whh, bih, bhh, h_all, out_hT, out_cT);
  k_head<<<N_, 128, 0, stream>>>(mask, out_sd, h_all, semb, lin_W, lin_b,
      lin2_W, lin2_b, out_y);
}